// HungarianMatcher_43327630082420
// MI455X (gfx1250) — compile-verified
//
#include <hip/hip_runtime.h>

// ---------------------------------------------------------------------------
// HungarianMatcher for MI455X (gfx1250)
//   Pass 1: fused cost-matrix build via V_WMMA_F32_16X16X4_F32
//           (rank-2 rotated-coordinate identity for 2D L1 distance)
//   Pass 2: LDS-resident auction assignment, one workgroup per batch,
//           deterministic winner selection via ds u64 atomic max.
// ---------------------------------------------------------------------------

typedef __attribute__((ext_vector_type(2))) float v2f;
typedef __attribute__((ext_vector_type(8))) float v8f;

#define NQ 2048
#define NG 512
#define QT (NQ / 16)   // 128 tiles along q
#define GT (NG / 16)   // 32 tiles along g

// ---------------------------------------------------------------------------
// Kernel 1: cost tiles. 8 waves per block, one 16(g) x 16(q) tile per wave.
//   CT[b][g][q] = 5 * max(|u_g-u_q|, |w_g-w_q|) + clamp(softplus(-conf_q))
// where u = x+y, w = x-y  (exact L1 identity in 2D).
// ---------------------------------------------------------------------------
__global__ __launch_bounds__(256) void hm_cost_kernel(
    const float* __restrict__ positions,    // [B, NQ, 2]
    const float* __restrict__ confidences,  // [B, NQ]
    const float* __restrict__ gt_positions, // [B, NG, 2]
    float* __restrict__ CT)                 // [B, NG, NQ]
{
    const int lane = threadIdx.x & 31;
    const int wave = threadIdx.x >> 5;
    const int tile = blockIdx.x * 8 + wave;

    const int g_tile = tile % GT;
    const int q_tile = (tile / GT) % QT;
    const int b      = tile / (GT * QT);

    const int gbase = g_tile * 16;
    const int qbase = q_tile * 16;
    const int l15   = lane & 15;
    const bool low  = lane < 16;      // lanes 16..31 carry K=2,3 (zeroed)

    // ---- A operand: gt side, M = g.  Row g = [u_g, 1, 0, 0] over K. ----
    const int   g  = gbase + l15;
    const float gx = gt_positions[((size_t)b * NG + g) * 2 + 0];
    const float gy = gt_positions[((size_t)b * NG + g) * 2 + 1];
    const float ug = gx + gy, wg = gx - gy;

    // ---- B operand: pred side, N = q.  Col q = [1, -u_q, 0, 0] over K. ----
    const int   q  = qbase + l15;
    const float px = positions[((size_t)b * NQ + q) * 2 + 0];
    const float py = positions[((size_t)b * NQ + q) * 2 + 1];
    const float uq = px + py, wq = px - py;

    // ISA 16x4 f32 A layout: VGPR0 = K0 (lanes 0-15) / K2 (lanes 16-31),
    //                        VGPR1 = K1 / K3.  B mirrors (rows across lanes).
    v2f aU, bU, aW, bW;
    aU.x = low ? ug   : 0.f;  aU.y = low ? 1.f : 0.f;
    bU.x = low ? 1.f  : 0.f;  bU.y = low ? -uq : 0.f;
    aW.x = low ? wg   : 0.f;  aW.y = low ? 1.f : 0.f;
    bW.x = low ? 1.f  : 0.f;  bW.y = low ? -wq : 0.f;

    v8f cz = {};
    // D = A x B + 0 : broadcast differences u_g - u_q / w_g - w_q per tile.
    v8f dU = __builtin_amdgcn_wmma_f32_16x16x4_f32(
        false, aU, false, bU, (short)0, cz, false, false);
    v8f dW = __builtin_amdgcn_wmma_f32_16x16x4_f32(
        false, aW, false, bW, (short)0, cz, false, false);

    // Confidence cost for this lane's column q:
    //   -log(clip(sigmoid(c), 1e-6, 1-1e-6)) == clamp(softplus(-c), lo, hi)
    const float c  = confidences[(size_t)b * NQ + q];
    float cc = __logf(1.f + __expf(-c));
    cc = fminf(fmaxf(cc, 1.0000005e-6f), 13.815511f);

    // C/D layout: VGPR r -> M = r (lanes 0-15) or r+8 (lanes 16-31), N = lane&15
    float* out = CT + (size_t)b * NG * NQ;
    const int grow = gbase + (low ? 0 : 8);
    const int qcol = qbase + l15;
#pragma unroll
    for (int r = 0; r < 8; ++r) {
        float cpos = fmaxf(fabsf(dU[r]), fabsf(dW[r]));
        out[(size_t)(grow + r) * NQ + qcol] = 5.0f * cpos + cc;
    }
}

// ---------------------------------------------------------------------------
// Kernel 2: auction assignment + output compaction. One block per batch,
// 512 threads = one thread per person (gt row). All state in LDS.
// ---------------------------------------------------------------------------
__global__ __launch_bounds__(512) void hm_auction_kernel(
    const float* __restrict__ CT,  // [B, NG, NQ]
    int* __restrict__ pred_out,    // [B, NG]
    int* __restrict__ gt_out)      // [B, NG]
{
    __shared__ float price[NQ];
    __shared__ int   owner[NQ];    // person owning object j, -1 = free
    __shared__ int   objOf[NG];    // object of person i, -1 = unassigned
    __shared__ unsigned long long bidkey[NQ];
    __shared__ int   scanbuf[NG];
    __shared__ int   nUn;

    const int tid = threadIdx.x;
    const int b   = blockIdx.x;
    const float* Crow = CT + ((size_t)b * NG + tid) * NQ;

    for (int j = tid; j < NQ; j += NG) price[j] = 0.f;

    // default-fill outputs so the buffer is fully initialized
    pred_out[(size_t)b * NG + tid] = 0;
    gt_out  [(size_t)b * NG + tid] = 0;

    const float epss[3] = {1.0f, 0.05f, 0.002f};
    for (int phase = 0; phase < 3; ++phase) {
        const float eps = epss[phase];
        for (int j = tid; j < NQ; j += NG) owner[j] = -1;
        objOf[tid] = -1;
        if (tid == 0) nUn = NG;
        __syncthreads();

        for (int iter = 0; iter < 768; ++iter) {
            for (int j = tid; j < NQ; j += NG) bidkey[j] = 0ull;
            __syncthreads();
            if (nUn == 0) break;   // uniform across block

            // ---- bidding: each unassigned person scans its cost row ----
            if (objOf[tid] < 0) {
                float v1 = -3.4e38f, v2 = -3.4e38f;
                int j1 = 0;
                for (int j = 0; j < NQ; ++j) {
                    float v = -Crow[j] - price[j];
                    if (v > v1) { v2 = v1; v1 = v; j1 = j; }
                    else if (v > v2) { v2 = v; }
                }
                float bid = price[j1] + (v1 - v2) + eps;  // > 0 always
                unsigned long long key =
                    ((unsigned long long)__float_as_uint(bid) << 32) |
                    (unsigned int)tid;
                atomicMax(&bidkey[j1], key);              // ds_max_u64
            }
            __syncthreads();

            // ---- assignment: highest deterministic bid wins object ----
            for (int j = tid; j < NQ; j += NG) {
                unsigned long long k = bidkey[j];
                if (k != 0ull) {
                    int person  = (int)(k & 0xffffffffull);
                    float bid   = __uint_as_float((unsigned int)(k >> 32));
                    int prev    = owner[j];
                    if (prev >= 0) { objOf[prev] = -1; atomicAdd(&nUn, 1); }
                    owner[j]      = person;
                    objOf[person] = j;
                    price[j]      = bid;
                    atomicAdd(&nUn, -1);
                }
            }
            __syncthreads();
        }
        __syncthreads();
    }

    // ---- compaction: emit (pred ascending, matched gt) ----
    int cnt = 0;
#pragma unroll
    for (int k = 0; k < NQ / NG; ++k)
        if (owner[tid * (NQ / NG) + k] >= 0) cnt++;
    scanbuf[tid] = cnt;
    __syncthreads();
    if (tid == 0) {
        int acc = 0;
        for (int i = 0; i < NG; ++i) { int t = scanbuf[i]; scanbuf[i] = acc; acc += t; }
    }
    __syncthreads();
    int pos = scanbuf[tid];
    for (int k = 0; k < NQ / NG; ++k) {
        int j = tid * (NQ / NG) + k;
        int p = owner[j];
        if (p >= 0 && pos < NG) {
            pred_out[(size_t)b * NG + pos] = j;
            gt_out  [(size_t)b * NG + pos] = p;
            pos++;
        }
    }
}

// ---------------------------------------------------------------------------
extern "C" void kernel_launch(void* const* d_in, const int* in_sizes, int n_in,
                              void* d_out, int out_size, void* d_ws, size_t ws_size,
                              hipStream_t stream) {
    const float* positions   = (const float*)d_in[0];  // [B, 2048, 2]
    const float* confidences = (const float*)d_in[1];  // [B, 2048]
    const float* gt_pos      = (const float*)d_in[2];  // [B, 512, 2]

    const int B = in_sizes[1] / NQ;

    float* CT = (float*)d_ws;                 // [B, 512, 2048] f32 = 67 MB
    int* pred_out = (int*)d_out;              // [B, 512]
    int* gt_out   = pred_out + (size_t)B * NG;

    const int tiles  = B * QT * GT;           // 16*128*32 = 65536
    const int blocks = tiles / 8;             // 8 waves (tiles) per block

    hm_cost_kernel<<<blocks, 256, 0, stream>>>(positions, confidences, gt_pos, CT);
    hm_auction_kernel<<<B, 512, 0, stream>>>(CT, pred_out, gt_out);
}